// NeuSFixedGridRenderer_60713657697174
// MI455X (gfx1250) — compile-verified
//
#include <hip/hip_runtime.h>
#include <math.h>

// ---------------------------------------------------------------------------
// NeuS fixed-grid renderer for MI455X (gfx1250, wave32).
// One wave32 per ray; 887 samples processed in chunks of 32 with a wave
// prefix-product for the transmittance cumprod; exact early-out when T==0.
// Final 32-lane x 9-channel reduction done with v_wmma_f32_16x16x4_f32
// (ones-matrix column sum -> layout-permutation invariant, hence correct).
// ---------------------------------------------------------------------------

typedef __attribute__((ext_vector_type(2))) float v2f;
typedef __attribute__((ext_vector_type(8))) float v8f;

#define N_SAMPLES 887
#define N_RAYS 4096
#define WAVES_PER_BLOCK 8
#define GRID_N 128
#define CH_STRIDE (128 * 128 * 128)

__device__ __forceinline__ float sigmoidf_fast(float x) {
    return 1.0f / (1.0f + __expf(-x));
}

__global__ __launch_bounds__(32 * WAVES_PER_BLOCK)
void neus_render_kernel(const float* __restrict__ sdf,   // 128^3
                        const float* __restrict__ inst,  // 8 x 128^3
                        const float* __restrict__ rays,  // 4096 x 8
                        const float* __restrict__ inv_s, // 1
                        float* __restrict__ out,         // 4096*8 inst + 4096 depth
                        float step) {
    __shared__ float lds[WAVES_PER_BLOCK][64];

    const int lane = threadIdx.x & 31;
    const int wave = threadIdx.x >> 5;
    const int ray  = blockIdx.x * WAVES_PER_BLOCK + wave;

    const float s = inv_s[0];
    const float ox = rays[ray * 8 + 0], oy = rays[ray * 8 + 1], oz = rays[ray * 8 + 2];
    const float dx = rays[ray * 8 + 3], dy = rays[ray * 8 + 4], dz = rays[ray * 8 + 5];
    const float nearv = rays[ray * 8 + 6], farv = rays[ray * 8 + 7];

    // ---- ray/box entry: t_min = clip(max(min(rate_a, rate_b)), near, far) ----
    const float vx = (dx == 0.f) ? 1e-6f : dx;
    const float vy = (dy == 0.f) ? 1e-6f : dy;
    const float vz = (dz == 0.f) ? 1e-6f : dz;
    const float tx = fminf((128.f - ox) / vx, -ox / vx);
    const float ty = fminf((128.f - oy) / vy, -oy / vy);
    const float tz = fminf((128.f - oz) / vz, -oz / vz);
    float tmin = fmaxf(fmaxf(tx, ty), tz);
    tmin = fminf(fmaxf(tmin, nearv), farv);

    const float S127 = 127.0f / 128.0f;  // (g+1)/2*(W-1) with g = p/64-1  ->  p*127/128
    float Tc = 1.0f;                     // running transmittance (wave-uniform)
    float acc[9];                        // 8 instance channels + depth
#pragma unroll
    for (int j = 0; j < 9; ++j) acc[j] = 0.f;

    for (int cb = 0; cb < N_SAMPLES; cb += 32) {
        const int i = cb + lane;
        const bool valid = (i < N_SAMPLES);
        const float z = tmin + step * (float)i;
        const float px = ox + dx * z, py = oy + dy * z, pz = oz + dz * z;
        const bool inbox = (px >= 0.f) & (px <= 128.f) &
                           (py >= 0.f) & (py <= 128.f) &
                           (pz >= 0.f) & (pz <= 128.f);

        // grid coords: vol[c, D, H, W]; D indexed by px, H by py, W by pz
        const float fx = fminf(fmaxf(px * S127, 0.f), 127.f);
        const float fy = fminf(fmaxf(py * S127, 0.f), 127.f);
        const float fz = fminf(fmaxf(pz * S127, 0.f), 127.f);
        const float x0f = floorf(fx), y0f = floorf(fy), z0f = floorf(fz);
        const int ix0 = (int)x0f, iy0 = (int)y0f, iz0 = (int)z0f;
        const float wx = fx - x0f, wy = fy - y0f, wz = fz - z0f;
        const int ix1 = min(ix0 + 1, 127), iy1 = min(iy0 + 1, 127), iz1 = min(iz0 + 1, 127);
        const int oX0 = ix0 << 14, oX1 = ix1 << 14;   // D stride = 128*128
        const int oY0 = iy0 << 7,  oY1 = iy1 << 7;    // H stride = 128
        const int b000 = oX0 + oY0 + iz0, b001 = oX0 + oY0 + iz1;
        const int b010 = oX0 + oY1 + iz0, b011 = oX0 + oY1 + iz1;
        const int b100 = oX1 + oY0 + iz0, b101 = oX1 + oY0 + iz1;
        const int b110 = oX1 + oY1 + iz0, b111 = oX1 + oY1 + iz1;

        // ---- SDF trilinear (8 gathers, L2-resident) ----
        const float s000 = sdf[b000], s001 = sdf[b001];
        const float s010 = sdf[b010], s011 = sdf[b011];
        const float s100 = sdf[b100], s101 = sdf[b101];
        const float s110 = sdf[b110], s111 = sdf[b111];
        const float wz0 = 1.f - wz, wy0 = 1.f - wy, wx0 = 1.f - wx;
        const float c00 = s000 * wz0 + s001 * wz;
        const float c01 = s010 * wz0 + s011 * wz;
        const float c10 = s100 * wz0 + s101 * wz;
        const float c11 = s110 * wz0 + s111 * wz;
        float sdfv = (c00 * wy0 + c01 * wy) * wx0 + (c10 * wy0 + c11 * wy) * wx;
        sdfv = inbox ? sdfv : 1000000.0f;

        const float dist = (i < N_SAMPLES - 1) ? step : 0.f;
        const float pc = sigmoidf_fast((sdfv + dist * 0.5f) * s);
        const float nc = sigmoidf_fast((sdfv - dist * 0.5f) * s);
        float alpha = (pc - nc + 1e-5f) / (pc + 1e-5f);
        alpha = fminf(fmaxf(alpha, 0.f), 1.f);
        if (!valid) alpha = 0.f;
        const float f = 1.f - alpha + 1e-10f;  // == 1.0f exactly for invalid lanes

        // ---- wave inclusive prefix product of f (5-step Hillis-Steele) ----
        float P = f;
#pragma unroll
        for (int off = 1; off < 32; off <<= 1) {
            const float v = __shfl_up(P, off, 32);
            if (lane >= off) P *= v;
        }
        float excl = __shfl_up(P, 1, 32);
        if (lane == 0) excl = 1.f;

        const float w = alpha * Tc * excl;
        acc[8] += w * z;  // depth (reference sums over all samples, no threshold)

        if (w > 0.0001f) {  // WEIGHT_THRES gate on instance contribution
            const float w000 = wx0 * wy0 * wz0, w001 = wx0 * wy0 * wz;
            const float w010 = wx0 * wy  * wz0, w011 = wx0 * wy  * wz;
            const float w100 = wx  * wy0 * wz0, w101 = wx  * wy0 * wz;
            const float w110 = wx  * wy  * wz0, w111 = wx  * wy  * wz;
#pragma unroll
            for (int c = 0; c < 8; ++c) {
                const float* __restrict__ g = inst + c * CH_STRIDE;
                const float v = g[b000] * w000 + g[b001] * w001 +
                                g[b010] * w010 + g[b011] * w011 +
                                g[b100] * w100 + g[b101] * w101 +
                                g[b110] * w110 + g[b111] * w111;
                acc[c] += w * v;
            }
        }

        Tc *= __shfl(P, 31, 32);   // carry transmittance (wave-uniform)
        if (Tc == 0.f) break;      // remaining weights are exactly zero
    }

    // ---- epilogue: reduce 32 lanes x 9 channels ----
    // butterfly to 4 partials per channel (lane mod 4)
#pragma unroll
    for (int j = 0; j < 9; ++j) {
        acc[j] += __shfl_xor(acc[j], 16, 32);
        acc[j] += __shfl_xor(acc[j], 8, 32);
        acc[j] += __shfl_xor(acc[j], 4, 32);
    }

    // stage 4x16 B-matrix (rows = the 4 partials, cols = channels, rest zero)
    float* lw = lds[wave];
    lw[lane] = 0.f;
    lw[lane + 32] = 0.f;
    __syncthreads();
    if (lane < 4) {
#pragma unroll
        for (int j = 0; j < 9; ++j) lw[lane * 16 + j] = acc[j];
    }
    __syncthreads();

    // D = ones(16x4) * B(4x16): every row of D = column sums of B.
    // With A == 1 the result is invariant to the hardware's K-row placement.
    const int col = lane & 15;
    const int half = lane >> 4;
    v2f a; a[0] = 1.f; a[1] = 1.f;
    v2f b; b[0] = lw[half * 16 + col]; b[1] = lw[(2 + half) * 16 + col];
    v8f cacc = {};
    v8f d = __builtin_amdgcn_wmma_f32_16x16x4_f32(
        /*neg_a=*/false, a, /*neg_b=*/false, b,
        /*c_mod=*/(short)0, cacc, /*reuse_a=*/false, /*reuse_b=*/false);

    const float r = d[0];  // row 0 (lanes 0-15), col = lane
    if (lane < 8) {
        out[ray * 8 + lane] = r;                 // instances_map[ray][lane]
    } else if (lane == 8) {
        out[N_RAYS * 8 + ray] = r;               // depth_map[ray]
    }
}

extern "C" void kernel_launch(void* const* d_in, const int* in_sizes, int n_in,
                              void* d_out, int out_size, void* d_ws, size_t ws_size,
                              hipStream_t stream) {
    const float* sdf  = (const float*)d_in[0];  // (1,1,128,128,128)
    const float* inst = (const float*)d_in[1];  // (1,8,128,128,128)
    const float* rays = (const float*)d_in[2];  // (4096,8)
    const float* invs = (const float*)d_in[3];  // (1,)

    float* out = (float*)d_out;                 // 4096*8 + 4096 floats

    // STEP_SIZE = sqrt(3*128^2) / 887, computed in double like the reference
    const float step = (float)(sqrt(49152.0) / 887.0);

    dim3 grid(N_RAYS / WAVES_PER_BLOCK);   // 512 blocks
    dim3 block(32 * WAVES_PER_BLOCK);      // 256 threads = 8 wave32s
    hipLaunchKernelGGL(neus_render_kernel, grid, block, 0, stream,
                       sdf, inst, rays, invs, out, step);
}